// NegSamplerMiniBatch_72971494359375
// MI455X (gfx1250) — compile-verified
//
#include <hip/hip_runtime.h>

typedef __attribute__((ext_vector_type(2))) float v2f;
typedef __attribute__((ext_vector_type(4))) float v4f;
typedef __attribute__((ext_vector_type(8))) float v8f;

#define DIM 256
#define KC  256

// tie-break like jax.lax.top_k: larger value wins; on equal values, smaller index wins
__device__ __forceinline__ bool better(float av, int ai, float bv, int bi) {
  return (av > bv) || (av == bv && ai < bi);
}

// c2[k] = ||centroid_k||^2  (256 floats into d_ws)
__global__ void NegSampler_c2_kernel(const float* __restrict__ cent,
                                     float* __restrict__ c2) {
  int k = threadIdx.x;                       // 256 threads, one per centroid
  const v4f* row = (const v4f*)(cent + k * DIM);
  float s = 0.f;
#pragma unroll
  for (int i = 0; i < DIM / 4; ++i) {
    v4f v = row[i];
    s += v.x * v.x + v.y * v.y + v.z * v.z + v.w * v.w;
  }
  c2[k] = s;
}

// Fused  x·C^T  (f32 WMMA)  ->  per-row top-2 (c2 - 2*xc)  ->  gather centroid row
__launch_bounds__(128)
__global__ void NegSamplerMiniBatch_kernel(const float* __restrict__ emb,
                                           const float* __restrict__ cent,
                                           const float* __restrict__ c2,
                                           float* __restrict__ out) {
  __shared__ float ldsA[4][16 * DIM];        // 64 KB: one 16x256 f32 tile per wave

  const int lane = threadIdx.x & 31;
  const int w    = threadIdx.x >> 5;         // wave id within block (wave32)
  const int m16  = lane & 15;                // row (A) / col (B) within 16-tile
  const int hl   = lane >> 4;                // K-half select per WMMA layout
  const int rowBase = blockIdx.x * 64 + w * 16;

  // ---- stage A tile into LDS, fully coalesced (32 x float4 per lane) ----
  {
    const v4f* src = (const v4f*)emb + (size_t)rowBase * (DIM / 4);
    v4f* dst = (v4f*)&ldsA[w][0];
#pragma unroll
    for (int i = 0; i < 32; ++i)
      dst[i * 32 + lane] = src[i * 32 + lane];
    // same-wave DS write->read is in-order (DScnt); no cross-wave sharing
  }

  // running top-2 per lane for 8 row-slots (row = r + 8*hl), cols = this lane's subset
  float v1[8], v2a[8];
  int   i1[8], i2a[8];
#pragma unroll
  for (int r = 0; r < 8; ++r) {
    v1[r] = -__builtin_inff(); v2a[r] = -__builtin_inff();
    i1[r] = 0x7fffffff;        i2a[r] = 0x7fffffff;
  }

  const float* aBase = &ldsA[w][m16 * DIM + 2 * hl];

  for (int tile = 0; tile < 16; ++tile) {
    const int ncol = tile * 16 + m16;
    const float* bBase = cent + (size_t)ncol * DIM + 2 * hl;

    v8f acc = {0.f, 0.f, 0.f, 0.f, 0.f, 0.f, 0.f, 0.f};
#pragma unroll
    for (int kb = 0; kb < 64; ++kb) {        // K = 256 in steps of 4
      v2f a = *(const v2f*)(aBase + kb * 4); // ds_load_b64
      v2f b = *(const v2f*)(bBase + kb * 4); // global_load_b64 (L2-resident)
      acc = __builtin_amdgcn_wmma_f32_16x16x4_f32(
          /*neg_a=*/false, a, /*neg_b=*/false, b,
          /*c_mod=*/(short)0, acc, /*reuse_a=*/false, /*reuse_b=*/false);
    }

    const float c2v = c2[ncol];
#pragma unroll
    for (int r = 0; r < 8; ++r) {
      float s = __builtin_fmaf(-2.f, acc[r], c2v);   // rank key: c2 - 2*xc
      if (better(s, ncol, v1[r], i1[r])) {
        v2a[r] = v1[r]; i2a[r] = i1[r]; v1[r] = s; i1[r] = ncol;
      } else if (better(s, ncol, v2a[r], i2a[r])) {
        v2a[r] = s; i2a[r] = ncol;
      }
    }
  }

  // ---- butterfly-merge top-2 across the 16 lanes of each half ----
#pragma unroll
  for (int off = 1; off <= 8; off <<= 1) {
#pragma unroll
    for (int r = 0; r < 8; ++r) {
      float w1 = __shfl_xor(v1[r],  off, 32);
      int   j1 = __shfl_xor(i1[r],  off, 32);
      float w2 = __shfl_xor(v2a[r], off, 32);
      int   j2 = __shfl_xor(i2a[r], off, 32);
      if (better(w1, j1, v1[r], i1[r])) {
        if (better(v1[r], i1[r], w2, j2)) { v2a[r] = v1[r]; i2a[r] = i1[r]; }
        else                              { v2a[r] = w2;    i2a[r] = j2;    }
        v1[r] = w1; i1[r] = j1;
      } else if (better(w1, j1, v2a[r], i2a[r])) {
        v2a[r] = w1; i2a[r] = j1;
      }
    }
  }

  // ---- gather 2nd-farthest centroid row per output row, coalesced B128 stores ----
  const v4f* cent4 = (const v4f*)cent;
  v4f* out4 = (v4f*)out;
#pragma unroll
  for (int m = 0; m < 16; ++m) {
    int idx = __shfl(i2a[m & 7], (m >> 3) * 16, 32);  // broadcast from owning half
    const v4f* srow = cent4 + (size_t)idx * (DIM / 4);
    v4f* drow = out4 + (size_t)(rowBase + m) * (DIM / 4);
    drow[lane]      = srow[lane];
    drow[lane + 32] = srow[lane + 32];
  }
}

extern "C" void kernel_launch(void* const* d_in, const int* in_sizes, int n_in,
                              void* d_out, int out_size, void* d_ws, size_t ws_size,
                              hipStream_t stream) {
  const float* emb  = (const float*)d_in[0];   // (262144, 256) f32
  const float* cent = (const float*)d_in[1];   // (256, 256) f32
  float* out = (float*)d_out;                  // (262144, 256) f32
  float* c2  = (float*)d_ws;                   // 256 f32 scratch

  NegSampler_c2_kernel<<<1, 256, 0, stream>>>(cent, c2);

  const int nRows = in_sizes[0] / DIM;         // 262144
  NegSamplerMiniBatch_kernel<<<nRows / 64, 128, 0, stream>>>(emb, cent, c2, out);
}